// VAERNN_53137335386739
// MI455X (gfx1250) — compile-verified
//
#include <hip/hip_runtime.h>
#include <math.h>

typedef __attribute__((ext_vector_type(2))) float v2f;
typedef __attribute__((ext_vector_type(8))) float v8f;

namespace {

constexpr int Lk = 256;           // sequence length
constexpr int Bk = 512;           // batch
constexpr int Rk = Lk * Bk;       // 131072 flattened rows

// ---- wave-level 16x16 f32 tile: C += A(16xK, row-major lda) * W^T(16 rows of K) ----
// Layout per CDNA5 ISA 7.12.2 (V_WMMA_F32_16X16X4_F32):
//   A frag: lane l -> M = l&15, regs hold K = 2*(l>>4)+{0,1}
//   B frag: lane l -> N = l&15, regs hold K = 2*(l>>4)+{0,1}  (B[k][n] == W[n][k])
//   C/D:    lane l -> N = l&15, VGPR v -> M = v + 8*(l>>4)
__device__ __forceinline__ v8f wmma16x16(const float* __restrict__ A, int lda,
                                         const float* __restrict__ W, int ldw,
                                         int K, v8f acc) {
  const int lane = threadIdx.x & 31;
  const int r  = lane & 15;
  const int h2 = (lane >> 4) << 1;           // 0 or 2
  const float* ap = A + r * lda + h2;
  const float* wp = W + r * ldw + h2;
  for (int k = 0; k < K; k += 4) {
    v2f a, b;
    a.x = ap[k]; a.y = ap[k + 1];
    b.x = wp[k]; b.y = wp[k + 1];
    acc = __builtin_amdgcn_wmma_f32_16x16x4_f32(false, a, false, b,
                                                (short)0, acc, false, false);
  }
  return acc;
}

__device__ __forceinline__ void store_tile(float* __restrict__ C, int ldc, v8f acc) {
  const int lane = threadIdx.x & 31;
  const int col = lane & 15;
  const int rb  = (lane >> 4) << 3;          // 0 or 8
#pragma unroll
  for (int v = 0; v < 8; ++v) C[(size_t)(rb + v) * ldc + col] = acc[v];
}

__device__ __forceinline__ v8f bias_init(const float* __restrict__ b) {
  const float bb = b[threadIdx.x & 15];
  v8f acc;
#pragma unroll
  for (int v = 0; v < 8; ++v) acc[v] = bb;
  return acc;
}

__device__ __forceinline__ float sigmoidf(float x) { return 1.0f / (1.0f + __expf(-x)); }

// ================= kernel 1: u-path preprocess + input-side GRU gates =================
// xg = relu(relu(X@w1^T+b1)@w2^T+b2) @ wih^T + bih        X:(R,64) -> xg:(R,384)
__global__ void __launch_bounds__(256)
k_pre_u(const float* __restrict__ X,
        const float* __restrict__ w1, const float* __restrict__ b1,    // (128,64)
        const float* __restrict__ w2, const float* __restrict__ b2,    // (128,128)
        const float* __restrict__ wih, const float* __restrict__ bih,  // (384,128)
        float* __restrict__ xg) {
  constexpr int LDH = 132;                   // 128 + 4 pad: kills LDS bank conflicts
  __shared__ float smem[64 * LDH];
  const int tid = threadIdx.x, wave = tid >> 5;
  const int row0 = blockIdx.x * 64;
  const int mt = wave >> 1;                  // 4 M-tiles, 2 waves each

  // stage X tile (64x64), coalesced
  for (int i = tid; i < 64 * 64; i += 256) {
    int rr = i >> 6, cc = i & 63;
    smem[rr * LDH + cc] = X[(size_t)(row0 + rr) * 64 + cc];
  }
  __syncthreads();

  v8f acc[4];
  // layer 1: 64x128 = 4x8 tiles, K=64
#pragma unroll
  for (int j = 0; j < 4; ++j) {
    const int nb = ((wave & 1) * 4 + j) * 16;
    v8f c = bias_init(b1 + nb);
    c = wmma16x16(smem + mt * 16 * LDH, LDH, w1 + nb * 64, 64, 64, c);
#pragma unroll
    for (int v = 0; v < 8; ++v) c[v] = fmaxf(c[v], 0.0f);
    acc[j] = c;
  }
  __syncthreads();
#pragma unroll
  for (int j = 0; j < 4; ++j)
    store_tile(smem + mt * 16 * LDH + ((wave & 1) * 4 + j) * 16, LDH, acc[j]);
  __syncthreads();

  // layer 2: 64x128, K=128
#pragma unroll
  for (int j = 0; j < 4; ++j) {
    const int nb = ((wave & 1) * 4 + j) * 16;
    v8f c = bias_init(b2 + nb);
    c = wmma16x16(smem + mt * 16 * LDH, LDH, w2 + nb * 128, 128, 128, c);
#pragma unroll
    for (int v = 0; v < 8; ++v) c[v] = fmaxf(c[v], 0.0f);
    acc[j] = c;
  }
  __syncthreads();
#pragma unroll
  for (int j = 0; j < 4; ++j)
    store_tile(smem + mt * 16 * LDH + ((wave & 1) * 4 + j) * 16, LDH, acc[j]);
  __syncthreads();

  // gates: 64x384 = 4x24 tiles (12 per wave), K=128, no activation
  for (int j = 0; j < 12; ++j) {
    const int nb = ((wave & 1) * 12 + j) * 16;
    v8f c = bias_init(bih + nb);
    c = wmma16x16(smem + mt * 16 * LDH, LDH, wih + nb * 128, 128, 128, c);
    store_tile(xg + (size_t)(row0 + mt * 16) * 384 + nb, 384, c);
  }
}

// ================= kernel 2: x-path preprocess -> x_embed (d_out slot 4) =============
__global__ void __launch_bounds__(256)
k_pre_x(const float* __restrict__ X,                                    // (R,128)
        const float* __restrict__ w1, const float* __restrict__ b1,     // (128,128)
        const float* __restrict__ w2, const float* __restrict__ b2,     // (128,128)
        float* __restrict__ out) {                                      // (R,128)
  constexpr int LDH = 132;
  __shared__ float smem[64 * LDH];
  const int tid = threadIdx.x, wave = tid >> 5;
  const int row0 = blockIdx.x * 64;
  const int mt = wave >> 1;

  for (int i = tid; i < 64 * 128; i += 256) {
    int rr = i >> 7, cc = i & 127;
    smem[rr * LDH + cc] = X[(size_t)(row0 + rr) * 128 + cc];
  }
  __syncthreads();

  v8f acc[4];
#pragma unroll
  for (int j = 0; j < 4; ++j) {
    const int nb = ((wave & 1) * 4 + j) * 16;
    v8f c = bias_init(b1 + nb);
    c = wmma16x16(smem + mt * 16 * LDH, LDH, w1 + nb * 128, 128, 128, c);
#pragma unroll
    for (int v = 0; v < 8; ++v) c[v] = fmaxf(c[v], 0.0f);
    acc[j] = c;
  }
  __syncthreads();
#pragma unroll
  for (int j = 0; j < 4; ++j)
    store_tile(smem + mt * 16 * LDH + ((wave & 1) * 4 + j) * 16, LDH, acc[j]);
  __syncthreads();

#pragma unroll
  for (int j = 0; j < 4; ++j) {
    const int nb = ((wave & 1) * 4 + j) * 16;
    v8f c = bias_init(b2 + nb);
    c = wmma16x16(smem + mt * 16 * LDH, LDH, w2 + nb * 128, 128, 128, c);
#pragma unroll
    for (int v = 0; v < 8; ++v) c[v] = fmaxf(c[v], 0.0f);
    store_tile(out + (size_t)(row0 + mt * 16) * 128 + nb, 128, c);
  }
}

// ================= kernel 3: GRU scan (parallel over batch rows) =====================
// Emits PRE-update hidden states h_seq[t] = h_{t-1} (h0 = 0), per reference scan.
__global__ void __launch_bounds__(256)
k_gru(const float* __restrict__ xg,     // (L,B,384)
      const float* __restrict__ whh,    // (384,128), L2-resident (192 KB)
      const float* __restrict__ bhh,    // (384)
      float* __restrict__ h_seq) {      // (L,B,128)
  constexpr int LDH = 132, LDG = 388;
  __shared__ float hbuf[16 * LDH];
  __shared__ float gbuf[16 * LDG];
  const int tid = threadIdx.x, wave = tid >> 5;
  const int brow0 = blockIdx.x * 16;

  for (int i = tid; i < 16 * LDH; i += 256) hbuf[i] = 0.0f;
  __syncthreads();

  for (int t = 0; t < Lk; ++t) {
    // hg = h @ whh^T + bhh : 16x384 -> 24 tiles, 3 per wave, K=128
#pragma unroll
    for (int j = 0; j < 3; ++j) {
      const int nb = (wave * 3 + j) * 16;
      v8f c = bias_init(bhh + nb);
      c = wmma16x16(hbuf, LDH, whh + nb * 128, 128, 128, c);
      store_tile(gbuf + nb, LDG, c);
    }
    __syncthreads();

    const size_t grow = (size_t)t * Bk + brow0;
    for (int i = tid; i < 16 * 128; i += 256) {
      const int rr = i >> 7, c = i & 127;
      const size_t gi = (grow + rr) * 384;
      const float xr = xg[gi + c], xz = xg[gi + 128 + c], xn = xg[gi + 256 + c];
      const float hr = gbuf[rr * LDG + c], hz = gbuf[rr * LDG + 128 + c],
                  hn = gbuf[rr * LDG + 256 + c];
      const float ho = hbuf[rr * LDH + c];
      const float rg = sigmoidf(xr + hr);
      const float zg = sigmoidf(xz + hz);
      const float ng = tanhf(xn + rg * hn);
      h_seq[(grow + rr) * 128 + c] = ho;               // pre-update hidden
      hbuf[rr * LDH + c] = (1.0f - zg) * ng + zg * ho;
    }
    __syncthreads();
  }
}

// ================= kernel 4: DBlock posterior head ===================================
__global__ void __launch_bounds__(256)
k_post(const float* __restrict__ xe,    // (R,128)
       const float* __restrict__ hs,    // (R,128)
       const float* __restrict__ eps,   // (R,128)
       const float* __restrict__ w1, const float* __restrict__ b1,   // (256,256)
       const float* __restrict__ w2, const float* __restrict__ b2,   // (256,256)
       const float* __restrict__ wmu, const float* __restrict__ bmu, // (128,256)
       const float* __restrict__ wls, const float* __restrict__ bls, // (128,256)
       float* __restrict__ out_mu, float* __restrict__ out_ls, float* __restrict__ out_sp) {
  constexpr int LDP = 260;
  __shared__ float smem[16 * LDP];
  const int tid = threadIdx.x, wave = tid >> 5;
  const int row0 = blockIdx.x * 16;

  // stage pin = concat(x_embed, h_seq) : 16 x 256
  for (int i = tid; i < 16 * 256; i += 256) {
    const int rr = i >> 8, cc = i & 255;
    const size_t gr = (size_t)(row0 + rr) * 128;
    smem[rr * LDP + cc] = (cc < 128) ? xe[gr + cc] : hs[gr + cc - 128];
  }
  __syncthreads();

  // t = relu(pin@w1^T+b1) * sigmoid(pin@w2^T+b2) : 16x256 -> 16 tiles, 2 per wave
  v8f tv[2];
#pragma unroll
  for (int j = 0; j < 2; ++j) {
    const int nb = (wave * 2 + j) * 16;
    v8f c1 = bias_init(b1 + nb);
    c1 = wmma16x16(smem, LDP, w1 + nb * 256, 256, 256, c1);
    v8f c2 = bias_init(b2 + nb);
    c2 = wmma16x16(smem, LDP, w2 + nb * 256, 256, 256, c2);
#pragma unroll
    for (int v = 0; v < 8; ++v) c1[v] = fmaxf(c1[v], 0.0f) * sigmoidf(c2[v]);
    tv[j] = c1;
  }
  __syncthreads();
#pragma unroll
  for (int j = 0; j < 2; ++j) store_tile(smem + (wave * 2 + j) * 16, LDP, tv[j]);
  __syncthreads();

  // mu / logsigma : 16x128 each, wave handles one 16-wide N tile of both, K=256
  {
    const int nb = wave * 16;
    v8f cmu = bias_init(bmu + nb);
    cmu = wmma16x16(smem, LDP, wmu + nb * 256, 256, 256, cmu);
    v8f cls = bias_init(bls + nb);
    cls = wmma16x16(smem, LDP, wls + nb * 256, 256, 256, cls);
    const int lane = tid & 31;
    const int col = lane & 15, rb = (lane >> 4) << 3;
#pragma unroll
    for (int v = 0; v < 8; ++v) {
      const size_t o = (size_t)(row0 + rb + v) * 128 + nb + col;
      const float m = cmu[v], s = cls[v];
      out_mu[o] = m;
      out_ls[o] = s;
      out_sp[o] = m + __expf(s) * eps[o];
    }
  }
}

}  // anonymous namespace

extern "C" void kernel_launch(void* const* d_in, const int* in_sizes, int n_in,
                              void* d_out, int out_size, void* d_ws, size_t ws_size,
                              hipStream_t stream) {
  (void)in_sizes; (void)n_in; (void)out_size; (void)ws_size;
  const float* ext  = (const float*)d_in[0];
  const float* obs  = (const float*)d_in[1];
  const float* eps  = (const float*)d_in[2];
  const float* u_w1 = (const float*)d_in[3];  const float* u_b1 = (const float*)d_in[4];
  const float* u_w2 = (const float*)d_in[5];  const float* u_b2 = (const float*)d_in[6];
  const float* x_w1 = (const float*)d_in[7];  const float* x_b1 = (const float*)d_in[8];
  const float* x_w2 = (const float*)d_in[9];  const float* x_b2 = (const float*)d_in[10];
  const float* p_w1 = (const float*)d_in[11]; const float* p_b1 = (const float*)d_in[12];
  const float* p_w2 = (const float*)d_in[13]; const float* p_b2 = (const float*)d_in[14];
  const float* p_wmu = (const float*)d_in[15]; const float* p_bmu = (const float*)d_in[16];
  const float* p_wls = (const float*)d_in[17]; const float* p_bls = (const float*)d_in[18];
  const float* wih = (const float*)d_in[19];  const float* whh = (const float*)d_in[20];
  const float* bih = (const float*)d_in[21];  const float* bhh = (const float*)d_in[22];

  float* out = (float*)d_out;
  const size_t RL = (size_t)Rk * 128;
  float* out_mu = out;           float* out_ls = out + RL;
  float* out_sp = out + 2 * RL;  float* out_h  = out + 3 * RL;
  float* out_xe = out + 4 * RL;

  float* xg = (float*)d_ws;      // (R, 384) fp32 gate pre-activations, 192 MB

  k_pre_u<<<Rk / 64, 256, 0, stream>>>(ext, u_w1, u_b1, u_w2, u_b2, wih, bih, xg);
  k_pre_x<<<Rk / 64, 256, 0, stream>>>(obs, x_w1, x_b1, x_w2, x_b2, out_xe);
  k_gru<<<Bk / 16, 256, 0, stream>>>(xg, whh, bhh, out_h);
  k_post<<<Rk / 16, 256, 0, stream>>>(out_xe, out_h, eps, p_w1, p_b1, p_w2, p_b2,
                                      p_wmu, p_bmu, p_wls, p_bls,
                                      out_mu, out_ls, out_sp);
}